// BinaryMeanpass3d_22454089023954
// MI455X (gfx1250) — compile-verified
//
#include <hip/hip_runtime.h>
#include <cstdint>
#include <cstddef>

// Grid geometry (fixed by the reference: shape (1,1,96,128,128), z contiguous)
#define NXC 96
#define NYC 128
#define NZC 128
#define SXc (NYC * NZC)   // 16384 floats between x-neighbors
#define SYc (NZC)         // 128 floats between y-neighbors
#define NTOT (NXC * NYC * NZC)  // 1,572,864

#define TAU  0.1f
#define ICC  0.25f
#define CCF  4.0f

// Fast sigmoid: v_exp_f32 + v_add + v_rcp_f32 (~1 ulp; fine inside a damped
// contractive fixed point). Avoids the ~12-op IEEE divide expansion.
__device__ __forceinline__ float sigmoidf_(float x) {
    return __builtin_amdgcn_rcpf(1.0f + __expf(-x));
}

__device__ __forceinline__ float clipr_(float v, float r) {
    return fminf(fmaxf(v, -r), r);
}

// ---------------------------------------------------------------------------
// Aug-Lag init: u = sigmoid(d); ps=pt=px=py=pz=div=0; g = tau*(0+0-0-u*icc)
// ---------------------------------------------------------------------------
__global__ __launch_bounds__(256) void k_init(
    const float* __restrict__ d,
    float* __restrict__ u,  float* __restrict__ ps, float* __restrict__ pt,
    float* __restrict__ px, float* __restrict__ py, float* __restrict__ pz,
    float* __restrict__ dv, float* __restrict__ g)
{
    const int idx = blockIdx.x * 256 + threadIdx.x;
    const float u0 = sigmoidf_(d[idx]);
    u[idx]  = u0;
    ps[idx] = 0.0f; pt[idx] = 0.0f;
    px[idx] = 0.0f; py[idx] = 0.0f; pz[idx] = 0.0f;
    dv[idx] = 0.0f;
    g[idx]  = -TAU * ICC * u0;
}

// ---------------------------------------------------------------------------
// Aug-Lag flow update, float4 along z:
//   p* = clip(p* + (g[nbr+] - g), -r*, r*)
// ---------------------------------------------------------------------------
__global__ __launch_bounds__(256) void k_flow(
    const float* __restrict__ rx, const float* __restrict__ ry,
    const float* __restrict__ rz, const float* __restrict__ g,
    float* __restrict__ px, float* __restrict__ py, float* __restrict__ pz)
{
    const int t    = blockIdx.x * 256 + threadIdx.x;
    const int base = t << 2;                  // linear voxel index of element 0
    const int z0   = base & (NZC - 1);        // multiple of 4
    const int y    = (base >> 7) & (NYC - 1);
    const int x    = base >> 14;

    const float4 gc = *(const float4*)(g + base);

    // --- z direction: neighbors shift within the vector; edge uses g[base+4]
    {
        const float4 rzv = *(const float4*)(rz + base);
        float4 pzv = *(float4*)(pz + base);
        const bool  wvalid = (z0 < NZC - 4);  // element 3 has z<127 iff z0<124
        const float gzp3 = wvalid ? g[base + 4] : 0.0f;
        float4 res;
        res.x = clipr_(pzv.x + (gc.y - gc.x), rzv.x);
        res.y = clipr_(pzv.y + (gc.z - gc.y), rzv.y);
        res.z = clipr_(pzv.z + (gc.w - gc.z), rzv.z);
        res.w = wvalid ? clipr_(pzv.w + (gzp3 - gc.w), rzv.w) : pzv.w;
        *(float4*)(pz + base) = res;
    }
    // --- y direction (condition uniform across the vector)
    if (y < NYC - 1) {
        const float4 gyp = *(const float4*)(g + base + SYc);
        const float4 ryv = *(const float4*)(ry + base);
        float4 pyv = *(float4*)(py + base);
        pyv.x = clipr_(pyv.x + (gyp.x - gc.x), ryv.x);
        pyv.y = clipr_(pyv.y + (gyp.y - gc.y), ryv.y);
        pyv.z = clipr_(pyv.z + (gyp.z - gc.z), ryv.z);
        pyv.w = clipr_(pyv.w + (gyp.w - gc.w), ryv.w);
        *(float4*)(py + base) = pyv;
    }
    // --- x direction (condition uniform across the vector)
    if (x < NXC - 1) {
        const float4 gxp = *(const float4*)(g + base + SXc);
        const float4 rxv = *(const float4*)(rx + base);
        float4 pxv = *(float4*)(px + base);
        pxv.x = clipr_(pxv.x + (gxp.x - gc.x), rxv.x);
        pxv.y = clipr_(pxv.y + (gxp.y - gc.y), rxv.y);
        pxv.z = clipr_(pxv.z + (gxp.z - gc.z), rxv.z);
        pxv.w = clipr_(pxv.w + (gxp.w - gc.w), rxv.w);
        *(float4*)(px + base) = pxv;
    }
}

// ---------------------------------------------------------------------------
// Per-voxel aug-lag source/sink/label update (shared by the 4 vector lanes)
// ---------------------------------------------------------------------------
__device__ __forceinline__ void auglag_point_(
    float dd, float ptv, float uu, float dvv,
    float& psn, float& ptn, float& un, float& gn)
{
    const float cs = fmaxf(dd, 0.0f);
    const float ct = -fminf(dd, 0.0f);
    psn = fminf(ICC + ptv + dvv - uu * ICC, cs);
    ptn = fminf(psn - dvv + uu * ICC, ct);
    un  = uu - CCF * (dvv - psn + ptn);
    gn  = TAU * (dvv + ptn - psn - un * ICC);
}

// ---------------------------------------------------------------------------
// Aug-Lag divergence + updates, float4 along z; also precompute next g
// ---------------------------------------------------------------------------
__global__ __launch_bounds__(256) void k_div(
    const float* __restrict__ d,
    float* __restrict__ ps, float* __restrict__ pt,
    const float* __restrict__ px, const float* __restrict__ py,
    const float* __restrict__ pz,
    float* __restrict__ dvout, float* __restrict__ u, float* __restrict__ g)
{
    const int t    = blockIdx.x * 256 + threadIdx.x;
    const int base = t << 2;
    const int z0   = base & (NZC - 1);
    const int y    = (base >> 7) & (NYC - 1);
    const int x    = base >> 14;

    const float4 pxv = *(const float4*)(px + base);
    const float4 pyv = *(const float4*)(py + base);
    const float4 pzv = *(const float4*)(pz + base);

    float4 dv;
    dv.x = pxv.x + pyv.x + pzv.x;
    dv.y = pxv.y + pyv.y + pzv.y;
    dv.z = pxv.z + pyv.z + pzv.z;
    dv.w = pxv.w + pyv.w + pzv.w;

    if (x > 0) {
        const float4 pxm = *(const float4*)(px + base - SXc);
        dv.x -= pxm.x; dv.y -= pxm.y; dv.z -= pxm.z; dv.w -= pxm.w;
    }
    if (y > 0) {
        const float4 pym = *(const float4*)(py + base - SYc);
        dv.x -= pym.x; dv.y -= pym.y; dv.z -= pym.z; dv.w -= pym.w;
    }
    // z: element 0 needs pz[base-1] (only if z0>0); elements 1..3 shift in-vector
    if (z0 > 0) dv.x -= pz[base - 1];
    dv.y -= pzv.x; dv.z -= pzv.y; dv.w -= pzv.z;

    const float4 dd  = *(const float4*)(d  + base);
    const float4 ptv = *(const float4*)(pt + base);
    const float4 uv  = *(const float4*)(u  + base);

    float4 psn, ptn, un, gn;
    auglag_point_(dd.x, ptv.x, uv.x, dv.x, psn.x, ptn.x, un.x, gn.x);
    auglag_point_(dd.y, ptv.y, uv.y, dv.y, psn.y, ptn.y, un.y, gn.y);
    auglag_point_(dd.z, ptv.z, uv.z, dv.z, psn.z, ptn.z, un.z, gn.z);
    auglag_point_(dd.w, ptv.w, uv.w, dv.w, psn.w, ptn.w, un.w, gn.w);

    *(float4*)(ps    + base) = psn;
    *(float4*)(pt    + base) = ptn;
    *(float4*)(dvout + base) = dv;
    *(float4*)(u     + base) = un;
    *(float4*)(g     + base) = gn;
}

// ---------------------------------------------------------------------------
// m0 = clip(u, 0, 1) in place (float4)
// ---------------------------------------------------------------------------
__global__ __launch_bounds__(256) void k_clamp(float* __restrict__ u)
{
    const int base = (blockIdx.x * 256 + threadIdx.x) << 2;
    float4 v = *(float4*)(u + base);
    v.x = fminf(fmaxf(v.x, 0.0f), 1.0f);
    v.y = fminf(fmaxf(v.y, 0.0f), 1.0f);
    v.z = fminf(fmaxf(v.z, 0.0f), 1.0f);
    v.w = fminf(fmaxf(v.w, 0.0f), 1.0f);
    *(float4*)(u + base) = v;
}

// ---------------------------------------------------------------------------
// Mean-pass damped sweep (FINAL=false): out = 0.5*m + 0.5*sigmoid(E(m))
// Final energy (FINAL=true):            out = E(m)
//
// Block = 512 threads = 16 wave32s covering (1 x-slab, 16 y-rows, 128 z);
// each thread computes 4 contiguous z voxels (all global/LDS traffic b128).
// 50 m-rows (25.6KB LDS) are staged via the CDNA5 async global->LDS datapath
// (global_load_async_to_lds_b128, ASYNCcnt): 18 center rows + 16 rows each
// at x-1 / x+1.  Halo amortization: 12.5 B/voxel of m traffic (vs 14 before).
//
// LDS tile rows:  0..17  -> (x,   y0-1 .. y0+16)  [clamped]
//                18..33  -> (x-1, y0   .. y0+15)  [clamped]
//                34..49  -> (x+1, y0   .. y0+15)  [clamped]
// ---------------------------------------------------------------------------
template <bool FINAL>
__global__ __launch_bounds__(512) void k_sweep(
    const float* __restrict__ d,
    const float* __restrict__ rx, const float* __restrict__ ry,
    const float* __restrict__ rz,
    const float* __restrict__ m, float* __restrict__ out)
{
    __shared__ __align__(16) float tile[50][NZC];

    const int tid  = threadIdx.x;
    const int x    = blockIdx.y;        // 0..95
    const int y0   = blockIdx.x * 16;   // 0..112
    const int wave = tid >> 5;          // 0..15
    const int lane = tid & 31;          // 0..31

    // ---- async stage: each wave copies rows wave, wave+16, wave+32, [wave+48]
    for (int r = wave; r < 50; r += 16) {
        int xs, ys;
        if (r < 18) {                   // rows y0-1 .. y0+16 at x (clamped)
            xs = x;
            ys = y0 - 1 + r;
            ys = ys < 0 ? 0 : (ys > NYC - 1 ? NYC - 1 : ys);
        } else if (r < 34) {            // rows y0 .. y0+15 at x-1 (clamped)
            xs = (x > 0) ? x - 1 : 0;
            ys = y0 + (r - 18);
        } else {                        // rows y0 .. y0+15 at x+1 (clamped)
            xs = (x < NXC - 1) ? x + 1 : NXC - 1;
            ys = y0 + (r - 34);
        }
        const float* src = m + ((size_t)(xs * NYC + ys) << 7) + (lane << 2);
        const unsigned ldsoff = (unsigned)(uintptr_t)(&tile[r][lane << 2]);
        asm volatile("global_load_async_to_lds_b128 %0, %1, off"
                     :: "v"(ldsoff), "v"(src) : "memory");
    }
    asm volatile("s_wait_asynccnt 0x0" ::: "memory");
    __syncthreads();

    // ---- compute: thread (ty,lane) handles voxels (x, y0+ty, z0..z0+3) ----
    const int ty = wave;                // 0..15
    const int z0 = lane << 2;           // 0,4,...,124
    const int y  = y0 + ty;
    const int idx = ((x * NYC + y) << 7) + z0;

    const float4 mc  = *(const float4*)&tile[1 + ty][z0];
    const float4 mym = *(const float4*)&tile[ty][z0];        // weight 0 if y==0
    const float4 myp = *(const float4*)&tile[2 + ty][z0];    // weight 0 if y==127
    const float4 mxm = *(const float4*)&tile[18 + ty][z0];   // weight 0 if x==0
    const float4 mxp = *(const float4*)&tile[34 + ty][z0];   // weight 0 if x==95
    const float mzm0 = tile[1 + ty][(z0 > 0) ? z0 - 1 : 0];          // weight 0 if z0==0
    const float mzp3 = tile[1 + ty][(z0 < NZC - 4) ? z0 + 4 : 127];  // weight 0 if z0==124

    float4 mzm; mzm.x = mzm0; mzm.y = mc.x; mzm.z = mc.y; mzm.w = mc.z;
    float4 mzp; mzp.x = mc.y; mzp.y = mc.z; mzp.z = mc.w; mzp.w = mzp3;

    const float4 zero4 = make_float4(0.f, 0.f, 0.f, 0.f);
    const float4 d4  = *(const float4*)(d + idx);
    float4 rxf = *(const float4*)(rx + idx);                 // address always valid
    if (x == NXC - 1) rxf = zero4;
    const float4 rxm = (x > 0) ? *(const float4*)(rx + idx - SXc) : zero4;
    float4 ryf = *(const float4*)(ry + idx);                 // address always valid
    if (y == NYC - 1) ryf = zero4;
    const float4 rym = (y > 0) ? *(const float4*)(ry + idx - SYc) : zero4;
    const float4 rz4 = *(const float4*)(rz + idx);
    float4 rzf = rz4;
    if (z0 == NZC - 4) rzf.w = 0.0f;                         // z==127 has no z+ term
    float4 rzm;
    rzm.x = (z0 > 0) ? rz[idx - 1] : 0.0f;
    rzm.y = rz4.x; rzm.z = rz4.y; rzm.w = rz4.z;

    float4 e = d4;
    e.x = fmaf(rxf.x, 2.0f * mxp.x - 1.0f, e.x);
    e.y = fmaf(rxf.y, 2.0f * mxp.y - 1.0f, e.y);
    e.z = fmaf(rxf.z, 2.0f * mxp.z - 1.0f, e.z);
    e.w = fmaf(rxf.w, 2.0f * mxp.w - 1.0f, e.w);
    e.x = fmaf(rxm.x, 2.0f * mxm.x - 1.0f, e.x);
    e.y = fmaf(rxm.y, 2.0f * mxm.y - 1.0f, e.y);
    e.z = fmaf(rxm.z, 2.0f * mxm.z - 1.0f, e.z);
    e.w = fmaf(rxm.w, 2.0f * mxm.w - 1.0f, e.w);
    e.x = fmaf(ryf.x, 2.0f * myp.x - 1.0f, e.x);
    e.y = fmaf(ryf.y, 2.0f * myp.y - 1.0f, e.y);
    e.z = fmaf(ryf.z, 2.0f * myp.z - 1.0f, e.z);
    e.w = fmaf(ryf.w, 2.0f * myp.w - 1.0f, e.w);
    e.x = fmaf(rym.x, 2.0f * mym.x - 1.0f, e.x);
    e.y = fmaf(rym.y, 2.0f * mym.y - 1.0f, e.y);
    e.z = fmaf(rym.z, 2.0f * mym.z - 1.0f, e.z);
    e.w = fmaf(rym.w, 2.0f * mym.w - 1.0f, e.w);
    e.x = fmaf(rzf.x, 2.0f * mzp.x - 1.0f, e.x);
    e.y = fmaf(rzf.y, 2.0f * mzp.y - 1.0f, e.y);
    e.z = fmaf(rzf.z, 2.0f * mzp.z - 1.0f, e.z);
    e.w = fmaf(rzf.w, 2.0f * mzp.w - 1.0f, e.w);
    e.x = fmaf(rzm.x, 2.0f * mzm.x - 1.0f, e.x);
    e.y = fmaf(rzm.y, 2.0f * mzm.y - 1.0f, e.y);
    e.z = fmaf(rzm.z, 2.0f * mzm.z - 1.0f, e.z);
    e.w = fmaf(rzm.w, 2.0f * mzm.w - 1.0f, e.w);

    float4 res;
    if (FINAL) {
        res = e;
    } else {
        res.x = fmaf(0.5f, sigmoidf_(e.x), 0.5f * mc.x);
        res.y = fmaf(0.5f, sigmoidf_(e.y), 0.5f * mc.y);
        res.z = fmaf(0.5f, sigmoidf_(e.z), 0.5f * mc.z);
        res.w = fmaf(0.5f, sigmoidf_(e.w), 0.5f * mc.w);
    }
    *(float4*)(out + idx) = res;
}

// ---------------------------------------------------------------------------
// Host-side orchestration (all on `stream`, graph-capture safe, fixed trip
// counts => deterministic).
// ---------------------------------------------------------------------------
extern "C" void kernel_launch(void* const* d_in, const int* in_sizes, int n_in,
                              void* d_out, int out_size, void* d_ws, size_t ws_size,
                              hipStream_t stream)
{
    const float* d  = (const float*)d_in[0];
    const float* rx = (const float*)d_in[1];
    const float* ry = (const float*)d_in[2];
    const float* rz = (const float*)d_in[3];
    float* out = (float*)d_out;

    // Workspace layout: 8 arrays of NTOT floats (~50 MB)
    float* ws = (float*)d_ws;
    float* u  = ws + (size_t)0 * NTOT;
    float* ps = ws + (size_t)1 * NTOT;
    float* pt = ws + (size_t)2 * NTOT;
    float* px = ws + (size_t)3 * NTOT;
    float* py = ws + (size_t)4 * NTOT;
    float* pz = ws + (size_t)5 * NTOT;
    float* dv = ws + (size_t)6 * NTOT;
    float* g  = ws + (size_t)7 * NTOT;

    const dim3 blk256(256);
    const dim3 blk512(512);
    const dim3 grd1(NTOT / 256);            // scalar pointwise (init)
    const dim3 grd4(NTOT / (256 * 4));      // float4 kernels: 1536 blocks
    const dim3 grds(NYC / 16, NXC);         // sweep tiles: (8, 96) = 768 blocks

    // ---- Aug-Lag max-flow: fixed 100 iterations (reference cap) ----
    k_init<<<grd1, blk256, 0, stream>>>(d, u, ps, pt, px, py, pz, dv, g);
    for (int it = 0; it < 100; ++it) {
        k_flow<<<grd4, blk256, 0, stream>>>(rx, ry, rz, g, px, py, pz);
        k_div <<<grd4, blk256, 0, stream>>>(d, ps, pt, px, py, pz, dv, u, g);
    }

    // ---- Mean-pass: m0 = clip(u,0,1); 2 rounds x 352 damped sweeps ----
    k_clamp<<<grd4, blk256, 0, stream>>>(u);
    float* mA = u;   // ping
    float* mB = ps;  // pong (aug-lag state no longer needed)
    const int inner_its = NXC + NYC + NZC;  // 352
    for (int s = 0; s < 2 * inner_its; ++s) {
        k_sweep<false><<<grds, blk512, 0, stream>>>(d, rx, ry, rz, mA, mB);
        float* t = mA; mA = mB; mB = t;
    }

    // ---- Final energy of converged m -> d_out ----
    k_sweep<true><<<grds, blk512, 0, stream>>>(d, rx, ry, rz, mA, out);
}